// Positional_GCN_32427003085126
// MI455X (gfx1250) — compile-verified
//
#include <hip/hip_runtime.h>

// ---------------------------------------------------------------------------
// Positional GCN (2x GCNConv with symmetric norm + self loops) for gfx1250.
// GEMMs use V_WMMA_F32_16X16X4_F32 (full fp32 matrix pipe, wave32 layout),
// with compile-time concat resolution and fully unrolled K loop.
// Aggregation uses per-edge float4 gathers + global f32 atomics (L2-resident:
// all intermediates fit in the 192MB L2).
// ---------------------------------------------------------------------------

typedef float v2f __attribute__((ext_vector_type(2)));
typedef float v8f __attribute__((ext_vector_type(8)));

__global__ void fill0_kernel(float* __restrict__ p, int n) {
  int i = blockIdx.x * blockDim.x + threadIdx.x;
  if (i < n) p[i] = 0.0f;
}

__global__ void count_deg_kernel(const int* __restrict__ dst, float* __restrict__ deg, int E) {
  int e = blockIdx.x * blockDim.x + threadIdx.x;
  if (e < E) atomicAdd(&deg[dst[e]], 1.0f);
}

__global__ void dinv_kernel(float* __restrict__ deg, int N) {
  int i = blockIdx.x * blockDim.x + threadIdx.x;
  if (i < N) deg[i] = rsqrtf(deg[i] + 1.0f);  // +1 for self loop
}

// H[N x F] = concat(A0[N x C0], A1[N x C1]) @ W[(C0+C1) x F]
// One wave32 per 16-row tile. A-matrix 16x4 f32 layout: lanes 0-15 hold
// K={kb,kb+1}, lanes 16-31 hold K={kb+2,kb+3}; B 4x16 mirrors with N across
// lanes; C/D 16x16: VGPR r -> row half*8+r, col = lane&15.
// C0 % 4 == 0 so each (k0,k0+1) pair is entirely in A0 or A1 -> the concat
// source is resolved at compile time and A loads are contiguous b64.
// Requires N % 16 == 0 (true: N = 100000).
template <int C0, int C1, int F>
__global__ void gemm_concat_wmma_kernel(const float* __restrict__ A0,
                                        const float* __restrict__ A1,
                                        const float* __restrict__ W,
                                        float* __restrict__ H, int N) {
  constexpr int K = C0 + C1;
  static_assert(C0 % 4 == 0 && C1 % 4 == 0 && F % 16 == 0, "layout");
  const int ntiles = N >> 4;
  const int wid  = (blockIdx.x * blockDim.x + threadIdx.x) >> 5;  // wave32
  const int lane = threadIdx.x & 31;
  if (wid >= ntiles) return;  // wave-uniform -> EXEC stays all-1s for WMMA

  const int half  = lane >> 4;   // 0: lanes 0-15, 1: lanes 16-31
  const int l15   = lane & 15;
  const int khalf = half * 2;    // K sub-offset within each step of 4
  const int arow  = wid * 16 + l15;

  // Preload the full A row fragment once (ct-invariant): K/4 x v2f = 32 VGPRs.
  const float* __restrict__ r0 = A0 + (size_t)arow * C0;
  const float* __restrict__ r1 = A1 + (size_t)arow * C1;
  v2f afrag[K / 4];
#pragma unroll
  for (int kb = 0; kb < K; kb += 4) {
    const int k0 = kb + khalf;  // even -> 8B-aligned v2f load
    if (kb < C0)
      afrag[kb / 4] = *(const v2f*)(r0 + k0);
    else
      afrag[kb / 4] = *(const v2f*)(r1 + (k0 - C0));
  }

#pragma unroll
  for (int ct = 0; ct < F; ct += 16) {
    const int col = ct + l15;
    v8f acc = {};
#pragma unroll
    for (int kb = 0; kb < K; kb += 4) {
      const int k0 = kb + khalf;
      v2f b;
      b.x = W[(size_t)k0 * F + col];
      b.y = W[(size_t)(k0 + 1) * F + col];
      acc = __builtin_amdgcn_wmma_f32_16x16x4_f32(
          /*neg_a=*/false, afrag[kb / 4], /*neg_b=*/false, b,
          /*c_mod=*/(short)0, acc, /*reuse_a=*/false, /*reuse_b=*/false);
    }
    float* __restrict__ out = H + (size_t)(wid * 16 + half * 8) * F + col;
#pragma unroll
    for (int r = 0; r < 8; ++r) out[(size_t)r * F] = acc[r];
  }
}

// One thread per (edge, 4-feature chunk): gather float4 of h[src], scale,
// scatter-add into agg[dst] with non-returning f32 atomics.
__global__ void scatter_kernel(const int* __restrict__ src, const int* __restrict__ dst,
                               const float* __restrict__ dinv,
                               const float* __restrict__ h,
                               float* __restrict__ agg, int F, int E) {
  int e = blockIdx.x * blockDim.x + threadIdx.x;
  if (e >= E) return;
  const int c = blockIdx.y << 2;
  const int s = src[e], d = dst[e];
  const float coef = dinv[s] * dinv[d];
  const float4 hv = *(const float4*)(h + (size_t)s * F + c);
  float* out = agg + (size_t)d * F + c;
  atomicAdd(out + 0, hv.x * coef);
  atomicAdd(out + 1, hv.y * coef);
  atomicAdd(out + 2, hv.z * coef);
  atomicAdd(out + 3, hv.w * coef);
}

// out = agg + h * dinv^2 (self loop) + bias, optional ReLU. In-place safe.
__global__ void finish_kernel(const float* __restrict__ agg, const float* __restrict__ h,
                              const float* __restrict__ dinv, const float* __restrict__ bias,
                              float* __restrict__ out, int F, int N, int do_relu) {
  int t = blockIdx.x * blockDim.x + threadIdx.x;
  if (t >= N * F) return;
  const int i = t / F;
  const int f = t - i * F;
  const float di = dinv[i];
  float v = agg[t] + h[t] * di * di + bias[f];
  out[t] = do_relu ? fmaxf(v, 0.0f) : v;
}

extern "C" void kernel_launch(void* const* d_in, const int* in_sizes, int n_in,
                              void* d_out, int out_size, void* d_ws, size_t ws_size,
                              hipStream_t stream) {
  (void)n_in; (void)out_size; (void)ws_size;

  const int*   edges = (const int*)d_in[0];   // [2, E]
  const float* feat  = (const float*)d_in[1]; // [N, 48]
  const float* loc   = (const float*)d_in[2]; // [N, 16]
  const float* W1    = (const float*)d_in[3]; // [64, 48] row-major
  const float* b1    = (const float*)d_in[4]; // [48]
  const float* W2    = (const float*)d_in[5]; // [64, 64]
  const float* b2    = (const float*)d_in[6]; // [64]

  constexpr int FRAW = 48, FLOC = 16, F1 = 48, FOUT = 64;
  const int E = in_sizes[0] / 2;
  const int N = in_sizes[1] / FRAW;
  const int* src = edges;
  const int* dst = edges + E;

  // Carve workspace (256B-aligned blocks): dinv | h1 | agg1 | h2  (~64.5MB)
  char* ws = (char*)d_ws;
  size_t off = 0;
  auto carve = [&](size_t bytes) -> char* {
    char* p = ws + off;
    off = (off + bytes + 255) & ~(size_t)255;
    return p;
  };
  float* dinv = (float*)carve((size_t)N * sizeof(float));
  float* h1   = (float*)carve((size_t)N * F1 * sizeof(float));
  float* agg1 = (float*)carve((size_t)N * F1 * sizeof(float));
  float* h2   = (float*)carve((size_t)N * FOUT * sizeof(float));
  float* outp = (float*)d_out;

  const int B = 256;
  // --- zero accumulators (d_out is poisoned by harness) ---
  fill0_kernel<<<(N + B - 1) / B, B, 0, stream>>>(dinv, N);
  fill0_kernel<<<(N * F1 + B - 1) / B, B, 0, stream>>>(agg1, N * F1);
  fill0_kernel<<<(N * FOUT + B - 1) / B, B, 0, stream>>>(outp, N * FOUT);

  // --- degrees -> dinv ---
  count_deg_kernel<<<(E + B - 1) / B, B, 0, stream>>>(dst, dinv, E);
  dinv_kernel<<<(N + B - 1) / B, B, 0, stream>>>(dinv, N);

  // --- layer 1: h1 = [feat | loc] @ W1 ---
  const int ntiles = N / 16;
  dim3 gblk(128);                       // 4 waves per block
  dim3 ggrd((ntiles + 3) / 4);
  gemm_concat_wmma_kernel<FRAW, FLOC, F1><<<ggrd, gblk, 0, stream>>>(feat, loc, W1, h1, N);

  dim3 sgrd1((E + B - 1) / B, F1 / 4);
  scatter_kernel<<<sgrd1, B, 0, stream>>>(src, dst, dinv, h1, agg1, F1, E);
  finish_kernel<<<(N * F1 + B - 1) / B, B, 0, stream>>>(agg1, h1, dinv, b1, agg1, F1, N, /*relu=*/1);

  // --- layer 2: h2 = [relu_out | loc] @ W2 ---
  gemm_concat_wmma_kernel<F1, FLOC, FOUT><<<ggrd, gblk, 0, stream>>>(agg1, loc, W2, h2, N);

  dim3 sgrd2((E + B - 1) / B, FOUT / 4);
  scatter_kernel<<<sgrd2, B, 0, stream>>>(src, dst, dinv, h2, outp, FOUT, E);
  finish_kernel<<<(N * FOUT + B - 1) / B, B, 0, stream>>>(outp, h2, dinv, b2, outp, FOUT, N, /*relu=*/0);
}